// PairwiseChannelAdd_54709293416528
// MI455X (gfx1250) — compile-verified
//
#include <hip/hip_runtime.h>
#include <hip/hip_bf16.h>

// PairwiseChannelAdd rewrite:
//   M[b,t,f]   = max_j x[b,j,t,f]
//   out[b,k,f] = max_t ( M[b,t,f] + x[b,k,t,f] ),  k in [0, C-2]
// Memory-bound fp32 streaming; WMMA is inapplicable (max-plus semiring), so the
// CDNA5 path exercised is the Tensor Data Mover (tensor_load_to_lds +
// s_wait_tensorcnt) double-buffered LDS pipeline in pass 2.

#define DEV __device__ __forceinline__

constexpr int Bn = 8, Cn = 160, Tn = 512, Fn = 32;
constexpr int TF   = Tn * Fn;         // 16384 floats per (b, channel) slice
constexpr int BTF  = Bn * Tn * Fn;    // 131072 floats in M
constexpr int JS   = 5;               // channel-split for pass 1 (occupancy)
constexpr int JCH  = Cn / JS;         // 32 channels per split
constexpr int CHUNK_T = 128;          // t-rows per TDM tile
constexpr int CHUNK_F = CHUNK_T * Fn; // 4096 floats = 16 KB per tile
constexpr int NCHUNK  = Tn / CHUNK_T; // 4 tiles per (b,k) slice

// ---------------- Tensor Data Mover (device-pass hard requirement) ----------
// Gated on __HIP_DEVICE_COMPILE__: the x86 host pass doesn't register AMDGCN
// builtins, so __has_builtin is false there (that's fine — device pass is what
// matters, and it must have the builtin or we refuse to build).

#if defined(__HIP_DEVICE_COMPILE__) && !__has_builtin(__builtin_amdgcn_tensor_load_to_lds)
#error "gfx1250 device pass: tensor_load_to_lds builtin not available"
#endif

typedef __attribute__((ext_vector_type(4))) unsigned int u32x4;
typedef __attribute__((ext_vector_type(8))) int i32x8;
typedef __attribute__((ext_vector_type(4))) int i32x4;

// 1-D tensor DMA: copy `nelem` fp32 from global `gsrc` into LDS byte offset
// `lds_byte_off`. Descriptor layout per CDNA5 ISA.
DEV void tdm_load_1d(const float* gsrc, unsigned lds_byte_off, unsigned nelem) {
#if defined(__HIP_DEVICE_COMPILE__)
  const unsigned long long ga = (unsigned long long)(uintptr_t)gsrc;
  u32x4 g0;
  g0[0] = 1u;                                        // count=1, user mode
  g0[1] = lds_byte_off;                              // lds_addr
  g0[2] = (unsigned)ga;                              // global_addr[31:0]
  g0[3] = (unsigned)((ga >> 32) & 0x01FFFFFFull)     // global_addr[56:32]
        | (2u << 30);                                // type=2 ("image")
  i32x8 g1;
  g1[0] = (int)(2u << 16);                           // data_size=4B; no pad/iter
  g1[1] = (int)(nelem << 16);                        // tensor_dim0[15:0]
  g1[2] = (int)((nelem >> 16) | (1u << 16));         // tensor_dim0[31:16] | tensor_dim1=1
  g1[3] = (int)(nelem << 16);                        // tile_dim0 (16-bit)
  g1[4] = 0;                                         // tile_dim1/2 unused
  g1[5] = (int)nelem;                                // tensor_dim0_stride[31:0]
  g1[6] = 0;
  g1[7] = 0;
  i32x4 z4 = {0, 0, 0, 0};
#if defined(__clang_major__) && (__clang_major__ >= 23)
  i32x8 z8 = {0, 0, 0, 0, 0, 0, 0, 0};
  __builtin_amdgcn_tensor_load_to_lds(g0, g1, z4, z4, z8, 0);
#else
  __builtin_amdgcn_tensor_load_to_lds(g0, g1, z4, z4, 0);
#endif
#else
  (void)gsrc; (void)lds_byte_off; (void)nelem;
#endif
}

DEV void wait_tensorcnt_le1() {
#if defined(__HIP_DEVICE_COMPILE__)
  __builtin_amdgcn_s_wait_tensorcnt(1);
#endif
}
DEV void wait_tensorcnt_le0() {
#if defined(__HIP_DEVICE_COMPILE__)
  __builtin_amdgcn_s_wait_tensorcnt(0);
#endif
}

// ---------------- Pass 2: out[b,k,f] = max_t (M + x[b,k]) -------------------
// (placed first in the file so the disasm snippet shows the TDM kernel)

__global__ void __launch_bounds__(256)
pca_maxplus_kernel(const float* __restrict__ x, const float* __restrict__ M,
                   float* __restrict__ out) {
  __shared__ float buf[2][CHUNK_F];   // 32 KB double buffer for x[b,k] tiles
  __shared__ float red[8][Fn];        // cross-wave reduction
  const int bk = blockIdx.x;          // one block per (b, k)
  const int b  = bk / (Cn - 1);
  const int k  = bk % (Cn - 1);
  const float* xk = x + ((size_t)b * Cn + k) * TF;
  const float* Mb = M + (size_t)b * TF;
  const int f  = threadIdx.x & 31;    // lane = feature -> coalesced M loads
  const int tw = threadIdx.x >> 5;    // wave index 0..7 = t-slice
  float acc = -__builtin_inff();

  // Wave 0 drives the Tensor Data Mover; two tiles in flight (TENSORcnt<=1).
  if (threadIdx.x < 32)
    tdm_load_1d(xk, (unsigned)(uintptr_t)&buf[0][0], CHUNK_F);
#pragma unroll
  for (int c = 0; c < NCHUNK; ++c) {
    if (threadIdx.x < 32) {
      if (c + 1 < NCHUNK) {
        tdm_load_1d(xk + (size_t)(c + 1) * CHUNK_F,
                    (unsigned)(uintptr_t)&buf[(c + 1) & 1][0], CHUNK_F);
        wait_tensorcnt_le1();                   // tile c landed (in-order)
      } else {
        wait_tensorcnt_le0();                   // last tile landed
      }
    }
    __syncthreads();                            // LDS tile visible to all waves
    const float* bp = &buf[c & 1][0];
    const int t0 = c * CHUNK_T;
#pragma unroll
    for (int i = 0; i < CHUNK_T / 8; ++i) {
      const int tt = tw + i * 8;
      acc = fmaxf(acc, bp[tt * Fn + f] + Mb[(size_t)(t0 + tt) * Fn + f]);
    }
    __syncthreads();                            // done reading before reuse
  }

  red[tw][f] = acc;
  __syncthreads();
  if (tw == 0) {
    float m = acc;
#pragma unroll
    for (int w = 1; w < 8; ++w) m = fmaxf(m, red[w][f]);
    out[((size_t)b * (Cn - 1) + k) * Fn + f] = m;  // [B,1,C-1,F] flat
  }
}

// ---------------- Pass 1: channel max -> M ----------------------------------

DEV float4 fmax4(float4 a, float4 b) {
  float4 r;
  r.x = fmaxf(a.x, b.x); r.y = fmaxf(a.y, b.y);
  r.z = fmaxf(a.z, b.z); r.w = fmaxf(a.w, b.w);
  return r;
}

__global__ void __launch_bounds__(256)
chanmax_partial_kernel(const float* __restrict__ x, float* __restrict__ part) {
  const int idx = blockIdx.x * 256 + threadIdx.x;  // JS * BTF/4 threads total
  const int js  = idx / (BTF / 4);
  const int e   = idx % (BTF / 4);                 // float4 index in [B,T,F]
  const int b   = e / (TF / 4);
  const int r   = e % (TF / 4);
  const float4* xv =
      (const float4*)x + ((size_t)b * Cn + (size_t)js * JCH) * (TF / 4) + r;
  float4 m = xv[0];
#pragma unroll 4
  for (int j = 1; j < JCH; ++j) m = fmax4(m, xv[(size_t)j * (TF / 4)]);
  ((float4*)part)[(size_t)js * (BTF / 4) + e] = m;
}

__global__ void __launch_bounds__(256)
chanmax_combine_kernel(const float* __restrict__ part, float* __restrict__ M) {
  const int e = blockIdx.x * 256 + threadIdx.x;    // BTF/4 threads total
  const float4* pv = (const float4*)part;
  float4 m = pv[e];
#pragma unroll
  for (int js = 1; js < JS; ++js) m = fmax4(m, pv[(size_t)js * (BTF / 4) + e]);
  ((float4*)M)[e] = m;
}

// ---------------- launch ----------------------------------------------------

extern "C" void kernel_launch(void* const* d_in, const int* in_sizes, int n_in,
                              void* d_out, int out_size, void* d_ws, size_t ws_size,
                              hipStream_t stream) {
  (void)in_sizes; (void)n_in; (void)out_size; (void)ws_size;
  const float* x = (const float*)d_in[0];
  float* out  = (float*)d_out;
  float* part = (float*)d_ws;                  // JS*BTF floats (2.5 MB)
  float* M    = part + (size_t)JS * BTF;       // BTF floats   (0.5 MB)

  chanmax_partial_kernel<<<(JS * BTF / 4) / 256, 256, 0, stream>>>(x, part);
  chanmax_combine_kernel<<<(BTF / 4) / 256, 256, 0, stream>>>(part, M);
  pca_maxplus_kernel<<<Bn * (Cn - 1), 256, 0, stream>>>(x, M, out);
}